// Quantizer_23776938950831
// MI455X (gfx1250) — compile-verified
//
#include <hip/hip_runtime.h>

typedef float v2f __attribute__((ext_vector_type(2)));
typedef float v8f __attribute__((ext_vector_type(8)));

#define TOKENS   16384
#define DIM      256
#define K_STEPS  64            // 256 / 4
#define NCODES   8192
#define NCHUNKS  512           // 8192 / 16
#define WPB      4             // waves per block
#define NBLOCKS  (TOKENS / 16) // 1024
#define LOSS_OFF 4194304       // 16384*256
#define IDX_OFF  4194305
#define LOSS_SCALE (1.25f / 4194304.f)

#define WMMA_F32(A, B, C) \
    __builtin_amdgcn_wmma_f32_16x16x4_f32(false, (A), false, (B), (short)0, (C), false, false)

// sched_group_barrier masks
#define SG_WMMA      0x008
#define SG_VMEM_READ 0x020

// ---------------- kernel 1: codebook row norms ----------------
__global__ __launch_bounds__(256) void vq_enorm_kernel(const float* __restrict__ emb,
                                                       float* __restrict__ enorm)
{
    int r = blockIdx.x * 256 + threadIdx.x;
    if (r < NCODES) {
        const float4* row = (const float4*)(emb + (size_t)r * DIM);
        float s = 0.f;
#pragma unroll 8
        for (int i = 0; i < DIM / 4; ++i) {
            float4 v = row[i];
            s += v.x * v.x + v.y * v.y + v.z * v.z + v.w * v.w;
        }
        enorm[r] = s;
    }
}

// ---------------- kernel 2: fused distance-GEMM + argmin + gather ----------------
__global__ __launch_bounds__(128, 1) void vq_main_kernel(const float* __restrict__ x,
                                                         const float* __restrict__ emb,
                                                         const float* __restrict__ enorm,
                                                         float* __restrict__ out,
                                                         float* __restrict__ loss_part)
{
    __shared__ float lds_d[WPB][16];
    __shared__ int   lds_n[WPB][16];
    __shared__ float lds_xn[16];
    __shared__ int   lds_fin[16];

    const int tid   = threadIdx.x;
    const int wave  = tid >> 5;
    const int lane  = tid & 31;
    const int l16   = lane & 15;
    const int hi    = lane >> 4;            // 0 or 1
    const int mbase = blockIdx.x * 16;

    // ---- A fragments: x-tile rows mbase..mbase+15, held in registers ----
    // ISA layout 32-bit A 16x4: lane m = l16, VGPR0 = K{0|2}, VGPR1 = K{1|3}
    const float* ap = x + (size_t)(mbase + l16) * DIM + 2 * hi;
    v2f a[K_STEPS];
#pragma unroll
    for (int k = 0; k < K_STEPS; ++k)
        a[k] = *(const v2f*)(ap + 4 * k);

    // xnorm[m]: per-lane partial sum-of-squares, combine the two halves
    {
        float s = 0.f;
#pragma unroll
        for (int k = 0; k < K_STEPS; ++k)
            s += a[k].x * a[k].x + a[k].y * a[k].y;
        s += __shfl_xor(s, 16, 32);
        if (wave == 0 && lane < 16) lds_xn[lane] = s;
    }

    float bestd[8];
    int   bestn[8];
#pragma unroll
    for (int r = 0; r < 8; ++r) { bestd[r] = 3.402823466e38f; bestn[r] = 0x7fffffff; }

    // ---- scan codebook: software-pipelined 8-deep load batches ----
    for (int chunk = wave; chunk < NCHUNKS; chunk += WPB) {
        const int n     = chunk * 16 + l16;
        const float* bp = emb + (size_t)n * DIM + 2 * hi;
        const float en  = enorm[n];

        v8f c = {};
        v2f b[2][8];                         // double-buffered batches

        // prime: batch 0 = k-steps 0..7
#pragma unroll
        for (int j = 0; j < 8; ++j)
            b[0][j] = *(const v2f*)(bp + 4 * j);

#pragma unroll
        for (int kb = 0; kb < 8; ++kb) {     // 8 batches x 8 k-steps
            const int cur = kb & 1;
            const int nxt = cur ^ 1;
            if (kb < 7) {
#pragma unroll
                for (int j = 0; j < 8; ++j)
                    b[nxt][j] = *(const v2f*)(bp + 4 * ((kb + 1) * 8 + j));
            }
#pragma unroll
            for (int j = 0; j < 8; ++j)
                c = WMMA_F32(a[kb * 8 + j], b[cur][j], c);
        }

        // ---- scheduler choreography: [L8] ([L8][W8])x7 [W8] ----
        __builtin_amdgcn_sched_group_barrier(SG_VMEM_READ, 8, 0);
#pragma unroll
        for (int kb = 0; kb < 7; ++kb) {
            __builtin_amdgcn_sched_group_barrier(SG_VMEM_READ, 8, 0);
            __builtin_amdgcn_sched_group_barrier(SG_WMMA, 8, 0);
        }
        __builtin_amdgcn_sched_group_barrier(SG_WMMA, 8, 0);

        // epilogue: dist2 - xnorm (xnorm row-constant; irrelevant for argmin)
#pragma unroll
        for (int r = 0; r < 8; ++r) {
            float d = en - 2.f * c[r];
            bool t = (d < bestd[r]) | ((d == bestd[r]) & (n < bestn[r]));
            bestd[r] = t ? d : bestd[r];
            bestn[r] = t ? n : bestn[r];
        }
    }

    // ---- reduce across the 16 lanes that share each m (per half-wave) ----
#pragma unroll
    for (int r = 0; r < 8; ++r) {
        float d = bestd[r];
        int   n = bestn[r];
#pragma unroll
        for (int off = 8; off >= 1; off >>= 1) {
            float od = __shfl_xor(d, off, 32);
            int   on = __shfl_xor(n, off, 32);
            bool t = (od < d) | ((od == d) & (on < n));
            d = t ? od : d;
            n = t ? on : n;
        }
        if (l16 == 0) {
            lds_d[wave][r + 8 * hi] = d;
            lds_n[wave][r + 8 * hi] = n;
        }
    }
    __syncthreads();

    // ---- cross-wave combine, index output, loss partial ----
    if (wave == 0) {
        float v = 0.f;
        if (lane < 16) {
            float d = lds_d[0][lane];
            int   n = lds_n[0][lane];
#pragma unroll
            for (int w = 1; w < WPB; ++w) {
                float od = lds_d[w][lane];
                int   on = lds_n[w][lane];
                bool t = (od < d) | ((od == d) & (on < n));
                d = t ? od : d;
                n = t ? on : n;
            }
            lds_fin[lane] = n;
            out[IDX_OFF + mbase + lane] = (float)n;  // indices as output dtype
            v = lds_xn[lane] + d;                    // full squared distance
        }
#pragma unroll
        for (int off = 16; off >= 1; off >>= 1)
            v += __shfl_xor(v, off, 32);
        if (lane == 0) loss_part[blockIdx.x] = v;
    }
    __syncthreads();

    // ---- gather winning codebook rows (straight-through fwd == quant) ----
    const float4* e4 = (const float4*)emb;
    float4*       o4 = (float4*)out;
    for (int i = tid; i < 16 * (DIM / 4); i += 128) {
        int m = i >> 6;
        int q = i & 63;
        int n = lds_fin[m];
        o4[(size_t)(mbase + m) * (DIM / 4) + q] = e4[(size_t)n * (DIM / 4) + q];
    }
}

// ---------------- kernel 3: deterministic loss reduction ----------------
__global__ __launch_bounds__(32) void vq_loss_kernel(const float* __restrict__ loss_part,
                                                     float* __restrict__ out)
{
    int lane = threadIdx.x;
    float v = 0.f;
    for (int i = lane; i < NBLOCKS; i += 32)
        v += loss_part[i];
#pragma unroll
    for (int off = 16; off >= 1; off >>= 1)
        v += __shfl_xor(v, off, 32);
    if (lane == 0) out[LOSS_OFF] = v * LOSS_SCALE;
}

extern "C" void kernel_launch(void* const* d_in, const int* in_sizes, int n_in,
                              void* d_out, int out_size, void* d_ws, size_t ws_size,
                              hipStream_t stream)
{
    const float* x   = (const float*)d_in[0];   // (4,4096,256) f32
    const float* emb = (const float*)d_in[1];   // (8192,256)   f32
    float* out = (float*)d_out;                 // [quant | loss | indices]

    float* enorm     = (float*)d_ws;            // 8192 floats
    float* loss_part = enorm + NCODES;          // 1024 floats

    vq_enorm_kernel<<<NCODES / 256, 256, 0, stream>>>(emb, enorm);
    vq_main_kernel<<<NBLOCKS, 128, 0, stream>>>(x, emb, enorm, out, loss_part);
    vq_loss_kernel<<<1, 32, 0, stream>>>(loss_part, out);
}